// GAT_9440338117264
// MI455X (gfx1250) — compile-verified
//
#include <hip/hip_runtime.h>
#include <math.h>

// GAT (2-layer, PyG-style) for MI455X gfx1250.
// Sparse phases: atomic-based segment max / sum / weighted scatter (memory/atomic bound).
// Dense phase (elu(out1+b1)@W2): V_WMMA_F32_16X16X4_F32, one wave per 16-node tile,
// branch-free inner loop (clamped global_load_b64 for A, padded LDS ds_load_b64 for B).

typedef __attribute__((ext_vector_type(2))) float v2f;
typedef __attribute__((ext_vector_type(8))) float v8f;

#define HEADS 8
#define HID   8
#define F1    64          // HEADS*HID
#define OUTC  10
#define NEG_SLOPE 0.2f

// branch-free leaky relu: for v>0 max(v,0.2v)=v; for v<=0 max(v,0.2v)=0.2v
__device__ __forceinline__ float leaky(float v) {
    return fmaxf(v, NEG_SLOPE * v);
}
// branch-free elu: v>0 -> v + (exp(0)-1) = v ; v<=0 -> 0 + exp(v)-1
__device__ __forceinline__ float elu1(float v) {
    return fmaxf(v, 0.0f) + __expf(fminf(v, 0.0f)) - 1.0f;
}
// float atomic max via sign-split int/uint atomics (monotone bit patterns).
__device__ __forceinline__ void atomicMaxF(float* addr, float v) {
    if (v >= 0.0f) atomicMax((int*)addr, __float_as_int(v));
    else           atomicMin((unsigned int*)addr, __float_as_uint(v));
}

// ---------------- init (runs every call: deterministic) ----------------
__global__ void k_init(float* out1, float* max1, float* den1, float* out2,
                       float* max2, float* den2, float* sums, float* cnt,
                       int N, int G) {
    long i = (long)blockIdx.x * blockDim.x + threadIdx.x;
    long N64 = (long)N * F1;
    long N8  = (long)N * 8;
    if (i >= N64) return;
    out1[i] = 0.0f;
    if (i < N8) { max1[i] = -INFINITY; den1[i] = 0.0f; out2[i] = 0.0f; }
    if (i < N)  { max2[i] = -INFINITY; den2[i] = 0.0f; }
    if (i < (long)G * 8) sums[i] = 0.0f;
    if (i < G) cnt[i] = 0.0f;
}

// ---------------- layer 1: per-node attention scalars ----------------
// IN==1 so h1[n,h,c] = x[n]*W1[h*8+c]; a_s1[n,h] = x[n] * <W1[h,:],att_src1[h,:]>
__global__ void k_node1(const float* __restrict__ x, const float* __restrict__ W1,
                        const float* __restrict__ att_s, const float* __restrict__ att_d,
                        float* a_s1, float* a_d1, int N) {
    int n = blockIdx.x * blockDim.x + threadIdx.x;
    if (n >= N) return;
    float xv = x[n];
#pragma unroll
    for (int h = 0; h < HEADS; ++h) {
        float s = 0.0f, d = 0.0f;
#pragma unroll
        for (int c = 0; c < HID; ++c) {
            float w = W1[h * HID + c];
            s += w * att_s[h * HID + c];
            d += w * att_d[h * HID + c];
        }
        a_s1[(long)n * HEADS + h] = xv * s;
        a_d1[(long)n * HEADS + h] = xv * d;
    }
}

__device__ __forceinline__ void msg_sd(const int* ei, long m, int E, int& src, int& dst) {
    if (m < E) { src = ei[m]; dst = ei[E + m]; }       // edge_index[0], edge_index[1]
    else       { src = dst = (int)(m - E); }           // self loop
}

// ---------------- layer 1 softmax: max, denom, aggregate ----------------
__global__ void k_max1(const int* __restrict__ ei, const float* __restrict__ a_s1,
                       const float* __restrict__ a_d1, float* max1, int E, int M) {
    long t = (long)blockIdx.x * blockDim.x + threadIdx.x;
    if (t >= (long)M * HEADS) return;
    int h = (int)(t & 7); long m = t >> 3;
    int src, dst; msg_sd(ei, m, E, src, dst);
    float e = leaky(a_s1[(long)src * HEADS + h] + a_d1[(long)dst * HEADS + h]);
    atomicMaxF(&max1[(long)dst * HEADS + h], e);
}

__global__ void k_den1(const int* __restrict__ ei, const float* __restrict__ a_s1,
                       const float* __restrict__ a_d1, const float* __restrict__ max1,
                       float* den1, int E, int M) {
    long t = (long)blockIdx.x * blockDim.x + threadIdx.x;
    if (t >= (long)M * HEADS) return;
    int h = (int)(t & 7); long m = t >> 3;
    int src, dst; msg_sd(ei, m, E, src, dst);
    long di = (long)dst * HEADS + h;
    float e = leaky(a_s1[(long)src * HEADS + h] + a_d1[di]);
    atomicAdd(&den1[di], __expf(e - max1[di]));
}

__global__ void k_agg1(const int* __restrict__ ei, const float* __restrict__ x,
                       const float* __restrict__ W1, const float* __restrict__ a_s1,
                       const float* __restrict__ a_d1, const float* __restrict__ max1,
                       const float* __restrict__ den1, float* out1, int E, int M) {
    long t = (long)blockIdx.x * blockDim.x + threadIdx.x;
    if (t >= (long)M * HEADS) return;
    int h = (int)(t & 7); long m = t >> 3;
    int src, dst; msg_sd(ei, m, E, src, dst);
    long di = (long)dst * HEADS + h;
    float e = leaky(a_s1[(long)src * HEADS + h] + a_d1[di]);
    float alpha = __expf(e - max1[di]) / (den1[di] + 1e-16f);
    float xs = x[src] * alpha;                 // h1[src,h,c] = x[src]*W1[h*8+c]
    float* o = out1 + (long)dst * F1 + h * HID;
#pragma unroll
    for (int c = 0; c < HID; ++c)
        atomicAdd(&o[c], xs * W1[h * HID + c]);
}

// ---------------- layer 2 projection: elu(out1+b1) @ W2 via WMMA f32 ----------------
// One wave per 16-node tile; K=64 in 16 steps of V_WMMA_F32_16X16X4_F32.
// A (16x4 f32): lane&15 = row M, lane>>4 selects K pair {2h,2h+1}.
// B (4x16 f32): lane&15 = col N, lane>>4 selects K pair. W2 padded 8->16 cols with 0,
//               staged column-major in LDS so each step is one ds_load_b64.
// D (16x16 f32): VGPR r -> M = r + 8*(lane>>4), N = lane&15.
__global__ void __launch_bounds__(256)
k_gemm_wmma(const float* __restrict__ out1, const float* __restrict__ b1,
            const float* __restrict__ W2, float* __restrict__ h2, int N) {
    __shared__ float sB[16 * F1];              // [col][k], cols 8..15 zero (4 KB)
    int t = threadIdx.x;
    for (int i = t; i < 16 * F1; i += 256) {
        int col = i >> 6, k = i & 63;
        sB[i] = (col < HID) ? W2[k * HID + col] : 0.0f;
    }
    __syncthreads();

    int wave = t >> 5, lane = t & 31;
    int half = lane >> 4, rc = lane & 15;
    long tile = (long)blockIdx.x * 8 + wave;   // 16 nodes per tile
    long nodeA = tile * 16 + rc;
    // Clamp for OOB safety: garbage A rows only affect D rows that are never stored.
    long nodeC = nodeA < (long)N ? nodeA : (long)N - 1;
    const float* arow = out1 + nodeC * F1;
    const float* brow = sB + rc * F1;

    v8f acc = {};
#pragma unroll
    for (int k0 = 0; k0 < 16; ++k0) {
        int k = k0 * 4 + half * 2;             // k is even -> 8B aligned
        float2 av = *(const float2*)(arow + k);    // global_load_b64
        float2 bv = *(const float2*)(brow + k);    // ds_load_b64
        float2 bb = *(const float2*)(b1 + k);
        v2f a, b;
        a[0] = elu1(av.x + bb.x);
        a[1] = elu1(av.y + bb.y);
        b[0] = bv.x;
        b[1] = bv.y;
        acc = __builtin_amdgcn_wmma_f32_16x16x4_f32(
            /*neg_a=*/false, a, /*neg_b=*/false, b,
            /*c_mod=*/(short)0, acc, /*reuse_a=*/false, /*reuse_b=*/false);
    }
#pragma unroll
    for (int r = 0; r < 8; ++r) {
        long n2 = tile * 16 + r + half * 8;
        if (n2 < N && rc < HID) h2[n2 * HID + rc] = acc[r];
    }
}

// ---------------- layer 2: per-node attention scalars ----------------
__global__ void k_node2(const float* __restrict__ h2, const float* __restrict__ att_s,
                        const float* __restrict__ att_d, float* a_s2, float* a_d2, int N) {
    int n = blockIdx.x * blockDim.x + threadIdx.x;
    if (n >= N) return;
    float s = 0.0f, d = 0.0f;
#pragma unroll
    for (int c = 0; c < HID; ++c) {
        float v = h2[(long)n * HID + c];
        s += v * att_s[c];
        d += v * att_d[c];
    }
    a_s2[n] = s; a_d2[n] = d;
}

__global__ void k_max2(const int* __restrict__ ei, const float* __restrict__ a_s2,
                       const float* __restrict__ a_d2, float* max2, int E, int M) {
    long m = (long)blockIdx.x * blockDim.x + threadIdx.x;
    if (m >= M) return;
    int src, dst; msg_sd(ei, m, E, src, dst);
    atomicMaxF(&max2[dst], leaky(a_s2[src] + a_d2[dst]));
}

__global__ void k_den2(const int* __restrict__ ei, const float* __restrict__ a_s2,
                       const float* __restrict__ a_d2, const float* __restrict__ max2,
                       float* den2, int E, int M) {
    long m = (long)blockIdx.x * blockDim.x + threadIdx.x;
    if (m >= M) return;
    int src, dst; msg_sd(ei, m, E, src, dst);
    float e = leaky(a_s2[src] + a_d2[dst]);
    atomicAdd(&den2[dst], __expf(e - max2[dst]));
}

__global__ void k_agg2(const int* __restrict__ ei, const float* __restrict__ h2,
                       const float* __restrict__ a_s2, const float* __restrict__ a_d2,
                       const float* __restrict__ max2, const float* __restrict__ den2,
                       float* out2, int E, int M) {
    long t = (long)blockIdx.x * blockDim.x + threadIdx.x;
    if (t >= (long)M * HID) return;
    int c = (int)(t & 7); long m = t >> 3;
    int src, dst; msg_sd(ei, m, E, src, dst);
    float e = leaky(a_s2[src] + a_d2[dst]);
    float alpha = __expf(e - max2[dst]) / (den2[dst] + 1e-16f);
    atomicAdd(&out2[(long)dst * HID + c], h2[(long)src * HID + c] * alpha);
}

// ---------------- global mean pool ----------------
__global__ void k_pool(const float* __restrict__ out2, const float* __restrict__ b2,
                       const int* __restrict__ batch, float* sums, float* cnt, int N) {
    int n = blockIdx.x * blockDim.x + threadIdx.x;
    if (n >= N) return;
    int g = batch[n];
#pragma unroll
    for (int c = 0; c < HID; ++c)
        atomicAdd(&sums[(long)g * HID + c], out2[(long)n * HID + c] + b2[c]);
    atomicAdd(&cnt[g], 1.0f);
}

// ---------------- final linear + log_softmax (512 threads; keep precise libm) -----
__global__ void k_final(const float* __restrict__ sums, const float* __restrict__ cnt,
                        const float* __restrict__ lin_w, const float* __restrict__ lin_b,
                        float* out, int G) {
    int g = blockIdx.x * blockDim.x + threadIdx.x;
    if (g >= G) return;
    float inv = 1.0f / fmaxf(cnt[g], 1.0f);
    float p[HID];
#pragma unroll
    for (int c = 0; c < HID; ++c) p[c] = sums[(long)g * HID + c] * inv;
    float logit[OUTC]; float mx = -INFINITY;
#pragma unroll
    for (int o = 0; o < OUTC; ++o) {
        float v = lin_b[o];
#pragma unroll
        for (int c = 0; c < HID; ++c) v += p[c] * lin_w[c * OUTC + o];
        logit[o] = v; mx = fmaxf(mx, v);
    }
    float s = 0.0f;
#pragma unroll
    for (int o = 0; o < OUTC; ++o) s += expf(logit[o] - mx);
    float lse = logf(s);
#pragma unroll
    for (int o = 0; o < OUTC; ++o) out[(long)g * OUTC + o] = logit[o] - mx - lse;
}

extern "C" void kernel_launch(void* const* d_in, const int* in_sizes, int n_in,
                              void* d_out, int out_size, void* d_ws, size_t ws_size,
                              hipStream_t stream) {
    const float* x     = (const float*)d_in[0];
    const int*   ei    = (const int*)  d_in[1];
    const int*   batch = (const int*)  d_in[2];
    // d_in[3] = num_graphs (device scalar); use out_size/10 host-side instead.
    const float* W1    = (const float*)d_in[4];
    const float* as1   = (const float*)d_in[5];
    const float* ad1   = (const float*)d_in[6];
    const float* b1    = (const float*)d_in[7];
    const float* W2    = (const float*)d_in[8];
    const float* as2   = (const float*)d_in[9];
    const float* ad2   = (const float*)d_in[10];
    const float* b2    = (const float*)d_in[11];
    const float* lw    = (const float*)d_in[12];
    const float* lb    = (const float*)d_in[13];
    float* out = (float*)d_out;

    const int N = in_sizes[0];          // x is [N,1]
    const int E = in_sizes[1] / 2;      // edge_index is [2,E]
    const int G = out_size / OUTC;
    const int M = E + N;                // messages incl. self loops

    // workspace carve-up (floats)
    float* ws   = (float*)d_ws;
    long N8 = (long)N * 8, N64 = (long)N * F1;
    float* a_s1 = ws;
    float* a_d1 = a_s1 + N8;
    float* max1 = a_d1 + N8;
    float* den1 = max1 + N8;
    float* out1 = den1 + N8;            // N*64
    float* h2   = out1 + N64;           // N*8
    float* a_s2 = h2   + N8;            // N
    float* a_d2 = a_s2 + N;
    float* max2 = a_d2 + N;
    float* den2 = max2 + N;
    float* out2 = den2 + N;             // N*8
    float* sums = out2 + N8;            // G*8
    float* cnt  = sums + (long)G * 8;   // G

    const int B = 256;
    long initN = N64;
    k_init<<<(int)((initN + B - 1) / B), B, 0, stream>>>(out1, max1, den1, out2,
                                                          max2, den2, sums, cnt, N, G);
    k_node1<<<(N + B - 1) / B, B, 0, stream>>>(x, W1, as1, ad1, a_s1, a_d1, N);

    long MH = (long)M * HEADS;
    int  gMH = (int)((MH + B - 1) / B);
    k_max1<<<gMH, B, 0, stream>>>(ei, a_s1, a_d1, max1, E, M);
    k_den1<<<gMH, B, 0, stream>>>(ei, a_s1, a_d1, max1, den1, E, M);
    k_agg1<<<gMH, B, 0, stream>>>(ei, x, W1, a_s1, a_d1, max1, den1, out1, E, M);

    int tiles = (N + 15) / 16;
    k_gemm_wmma<<<(tiles + 7) / 8, B, 0, stream>>>(out1, b1, W2, h2, N);

    k_node2<<<(N + B - 1) / B, B, 0, stream>>>(h2, as2, ad2, a_s2, a_d2, N);

    int gM = (M + B - 1) / B;
    k_max2<<<gM, B, 0, stream>>>(ei, a_s2, a_d2, max2, E, M);
    k_den2<<<gM, B, 0, stream>>>(ei, a_s2, a_d2, max2, den2, E, M);
    long MC = (long)M * HID;
    k_agg2<<<(int)((MC + B - 1) / B), B, 0, stream>>>(ei, h2, a_s2, a_d2, max2, den2,
                                                      out2, E, M);

    k_pool<<<(N + B - 1) / B, B, 0, stream>>>(out2, b2, batch, sums, cnt, N);
    k_final<<<(G + B - 1) / B, B, 0, stream>>>(sums, cnt, lw, lb, out, G);
}